// adviser_loss_49074296324779
// MI455X (gfx1250) — compile-verified
//
#include <hip/hip_runtime.h>
#include <math.h>

#define NBINS 37
#define ROWB  (NBINS * 4)   // 148 bytes per sample row
#define TILE  128           // samples per block tile (tile bytes = 18944, 16B-divisible)
#define BLK   128           // threads per block (4 waves of 32)
#define MAXNBLK 2048

#if defined(__has_builtin)
# if __has_builtin(__builtin_amdgcn_global_load_async_to_lds_b128) && \
     __has_builtin(__builtin_amdgcn_s_wait_asynccnt)
#  define USE_ASYNC 1
# endif
#endif
#ifndef USE_ASYNC
# define USE_ASYNC 0
#endif

typedef int v4i __attribute__((ext_vector_type(4)));
typedef __attribute__((address_space(1))) v4i glb_v4i_t;
typedef __attribute__((address_space(3))) v4i lds_v4i_t;

__global__ __launch_bounds__(BLK) void adviser_loss_partial_kernel(
    const float* __restrict__ preds, const float* __restrict__ labels,
    const int* __restrict__ obj, int batch, float* __restrict__ ws)
{
    __shared__ __align__(16) char smem[TILE * ROWB * 2];  // preds tile | labels tile
    __shared__ float red[3][BLK];

    const int tid = threadIdx.x;
    char* lP = smem;
    char* lL = smem + TILE * ROWB;

    float acc0 = 0.0f, acc1 = 0.0f, acc2 = 0.0f;

    const int nTiles = (batch + TILE - 1) / TILE;
    for (int tile = blockIdx.x; tile < nTiles; tile += gridDim.x) {
        const int base = tile * TILE;
        int rows = batch - base;
        if (rows > TILE) rows = TILE;
        // rows is always a multiple of 4 here (batch % 4 == 0), so chunk count is exact.
        const int nch = (rows * NBINS) / 4;  // 16-byte chunks per array

        const char* gP = (const char*)(preds  + (size_t)base * NBINS);
        const char* gL = (const char*)(labels + (size_t)base * NBINS);

#if USE_ASYNC
        for (int i = tid; i < nch; i += BLK) {
            __builtin_amdgcn_global_load_async_to_lds_b128(
                (glb_v4i_t*)(gP + (size_t)i * 16), (lds_v4i_t*)(lP + i * 16), 0, 0);
            __builtin_amdgcn_global_load_async_to_lds_b128(
                (glb_v4i_t*)(gL + (size_t)i * 16), (lds_v4i_t*)(lL + i * 16), 0, 0);
        }
        __builtin_amdgcn_s_wait_asynccnt(0);
#else
        for (int i = tid; i < nch; i += BLK) {
            float4 a = *(const float4*)(gP + (size_t)i * 16);
            float4 b = *(const float4*)(gL + (size_t)i * 16);
            *(float4*)(lP + i * 16) = a;
            *(float4*)(lL + i * 16) = b;
        }
#endif
        __syncthreads();

        if (tid < rows) {
            const int s     = base + tid;
            const int c     = obj[s];
            const int start = c * 13;                 // {0,13,26}
            const int len   = (c == 2) ? 11 : 13;     // ends {13,26,37}

            const float* P = (const float*)(lP + tid * ROWB);
            const float* L = (const float*)(lL + tid * ROWB);

            float pv[13], lv[13];
            float mp = -1e30f, ml = -1e30f;
            #pragma unroll
            for (int j = 0; j < 13; ++j) {
                int k = start + j;
                if (k > NBINS - 1) k = NBINS - 1;     // clamp (value masked below)
                const bool act = j < len;
                float a = P[k];
                float b = L[k];
                pv[j] = act ? a : -1e30f;             // reference's NEG_INF sentinel
                lv[j] = act ? b : -1e30f;
                mp = fmaxf(mp, pv[j]);
                ml = fmaxf(ml, lv[j]);
            }

            float sp = 0.0f, sl = 0.0f;
            #pragma unroll
            for (int j = 0; j < 13; ++j) {
                float e = __expf(pv[j] - mp);         // masked -> exp underflows to 0
                float f = __expf(lv[j] - ml);
                sp += e;
                lv[j] = f;                            // keep exp(labels) for t
                sl += f;
            }

            const float lse    = mp + __logf(sp);
            const float inv_sl = __builtin_amdgcn_rcpf(sl);

            float loss = 0.0f;
            #pragma unroll
            for (int j = 0; j < 13; ++j) {
                float logp   = pv[j] - lse;                       // masked: ~-1e30
                float p      = __expf(logp);                      // masked: 0
                float t      = lv[j] * inv_sl;                    // masked: 0
                float logp_c = fmaxf(logp, -100.0f);
                float l1mp   = fmaxf(__logf(1.0f - p), -100.0f);  // NaN/-inf -> -100
                // masked bins: t=0, l1mp=log(1)=0 -> term contributes exactly 0
                loss -= t * logp_c + (1.0f - t) * l1mp;
            }

            if (c == 0)      acc0 += loss;
            else if (c == 1) acc1 += loss;
            else             acc2 += loss;
        }
        __syncthreads();  // protect LDS tile before next iteration overwrites it
    }

    // Deterministic block tree reduction of the 3 class partials.
    red[0][tid] = acc0; red[1][tid] = acc1; red[2][tid] = acc2;
    __syncthreads();
    #pragma unroll
    for (int off = BLK / 2; off > 0; off >>= 1) {
        if (tid < off) {
            red[0][tid] += red[0][tid + off];
            red[1][tid] += red[1][tid + off];
            red[2][tid] += red[2][tid + off];
        }
        __syncthreads();
    }
    if (tid == 0) {
        ws[0 * gridDim.x + blockIdx.x] = red[0][0];
        ws[1 * gridDim.x + blockIdx.x] = red[1][0];
        ws[2 * gridDim.x + blockIdx.x] = red[2][0];
    }
}

__global__ __launch_bounds__(256) void adviser_loss_finalize_kernel(
    const float* __restrict__ ws, const float* __restrict__ w,
    float* __restrict__ out, int nblk, float batchf)
{
    __shared__ float red[3][256];
    const int tid = threadIdx.x;
    float s0 = 0.0f, s1 = 0.0f, s2 = 0.0f;
    for (int b = tid; b < nblk; b += 256) {
        s0 += ws[0 * nblk + b];
        s1 += ws[1 * nblk + b];
        s2 += ws[2 * nblk + b];
    }
    red[0][tid] = s0; red[1][tid] = s1; red[2][tid] = s2;
    __syncthreads();
    #pragma unroll
    for (int off = 128; off > 0; off >>= 1) {
        if (tid < off) {
            red[0][tid] += red[0][tid + off];
            red[1][tid] += red[1][tid + off];
            red[2][tid] += red[2][tid + off];
        }
        __syncthreads();
    }
    if (tid == 0) {
        float l0 = red[0][0] * w[0];
        float l1 = red[1][0] * w[1];
        float l2 = red[2][0] * w[2];
        out[1] = l0;
        out[2] = l1;
        out[3] = l2;
        out[0] = (l0 + l1 + l2) / batchf;
    }
}

extern "C" void kernel_launch(void* const* d_in, const int* in_sizes, int n_in,
                              void* d_out, int out_size, void* d_ws, size_t ws_size,
                              hipStream_t stream) {
    const float* preds   = (const float*)d_in[0];
    const float* labels  = (const float*)d_in[1];
    const float* weights = (const float*)d_in[2];
    const int*   obj     = (const int*)d_in[3];
    const int    batch   = in_sizes[3];

    float* out = (float*)d_out;
    float* ws  = (float*)d_ws;

    int nTiles = (batch + TILE - 1) / TILE;
    int nblk = MAXNBLK;
    if (nblk > nTiles) nblk = nTiles;
    size_t need = (size_t)nblk * 3 * sizeof(float);
    if (ws_size < need) {
        int m = (int)(ws_size / (3 * sizeof(float)));
        nblk = m > 0 ? m : 1;
    }

    adviser_loss_partial_kernel<<<nblk, BLK, 0, stream>>>(preds, labels, obj, batch, ws);
    adviser_loss_finalize_kernel<<<1, 256, 0, stream>>>(ws, weights, out, nblk, (float)batch);
}